// SparseStateAggregator_39573828665393
// MI455X (gfx1250) — compile-verified
//
#include <hip/hip_runtime.h>

typedef float         v2f __attribute__((ext_vector_type(2)));
typedef float         v8f __attribute__((ext_vector_type(8)));
typedef unsigned int  v4u __attribute__((ext_vector_type(4)));
typedef int           v4i __attribute__((ext_vector_type(4)));
typedef int           v8i __attribute__((ext_vector_type(8)));

namespace {
constexpr int   SEQ      = 4096;
constexpr int   DIM      = 1024;
constexpr int   MST      = 64;     // MAX_STATES
constexpr int   TBLK     = 8;      // tokens per block (padded to WMMA M=16)
constexpr int   NBLK     = SEQ / TBLK;
constexpr float THRESH_F = 0.9f;
constexpr float EPS_F    = 1e-12f;
constexpr int   NTHREADS = 512;    // 16 waves (wave32)

// ---- dynamic LDS layout (in floats) ----
constexpr int C_OFF    = 0;                       // centroids [64][1024]
constexpr int K_OFF    = C_OFF + MST * DIM;       // key tile  [8][1024] (TDM target)
constexpr int P_OFF    = K_OFF + TBLK * DIM;      // WMMA partials [16][32][8]
constexpr int DOT_OFF  = P_OFF + 16 * 32 * 8;     // dots [16][64]
constexpr int G_OFF    = DOT_OFF + 16 * 64;       // Gram [16][16]
constexpr int CNT_OFF  = G_OFF + 256;             // counts [64]
constexpr int NCS_OFF  = CNT_OFF + MST;           // ||c||^2 [64]
constexpr int IVC_OFF  = NCS_OFF + MST;           // rsqrt(||c||^2+eps) [64]
constexpr int SLOT_OFF = IVC_OFF + MST;           // slot per token [8] (int)
constexpr int DEN_OFF  = SLOT_OFF + TBLK;         // 1/denom per token [8]
constexpr int KK_OFF   = DEN_OFF + TBLK;          // K (num states), int
constexpr int LDS_FLOATS = KK_OFF + 8;
} // namespace

__global__ __launch_bounds__(NTHREADS)
void sparse_state_agg_kernel(const float* __restrict__ keys,
                             const float* __restrict__ vals,
                             float* __restrict__ out) {
  extern __shared__ float smem[];
  const int tid  = threadIdx.x;
  const int lane = tid & 31;
  const int wv   = tid >> 5;               // wave id 0..15
  const int b    = blockIdx.x;

  const float* kb = keys + (size_t)b * SEQ * DIM;
  const float* vb = vals + (size_t)b * SEQ * DIM;
  float*       ob = out  + (size_t)b * MST * DIM;   // states live here

  float* C    = smem + C_OFF;
  float* Kst  = smem + K_OFF;
  float* P    = smem + P_OFF;
  float* dots = smem + DOT_OFF;
  float* G    = smem + G_OFF;
  float* cnt  = smem + CNT_OFF;
  float* ncs  = smem + NCS_OFF;
  float* ivc  = smem + IVC_OFF;
  int*   slotA= (int*)(smem + SLOT_OFF);
  float* denA = smem + DEN_OFF;
  int*   Kp   = (int*)(smem + KK_OFF);

  const float NEGINF = -__builtin_inff();

  // ---- init: each thread owns columns {2*tid, 2*tid+1} of every row ----
  {
    const int j0 = tid * 2;
    for (int r = 0; r < MST; ++r) {
      C[r * DIM + j0]     = 0.0f;  C[r * DIM + j0 + 1]     = 0.0f;
      ob[r * DIM + j0]    = 0.0f;  ob[r * DIM + j0 + 1]    = 0.0f;
    }
    for (int i = tid; i < MST; i += NTHREADS) { cnt[i] = 0.0f; ncs[i] = 0.0f; ivc[i] = 0.0f; }
    if (tid == 0) *Kp = 0;
  }
  __syncthreads();

  for (int blk = 0; blk < NBLK; ++blk) {
    const int t0 = blk * TBLK;

    // ---- TDM: stage key tile [8 x 1024] f32 -> LDS (one wave issues) ----
    if (wv == 0) {
      unsigned long long ga = (unsigned long long)(const void*)(kb + (size_t)t0 * DIM);
      v4u g0;
      g0[0] = 1u;                                        // count=1, user mode
      g0[1] = (unsigned)(K_OFF * 4);                     // lds_addr (bytes)
      g0[2] = (unsigned)(ga & 0xFFFFFFFFu);              // global_addr lo
      g0[3] = (unsigned)((ga >> 32) & 0x01FFFFFFu) | (2u << 30); // addr hi | type=2
      v8i g1;
      g1[0] = (2 << 16);                                 // data_size=4B, no multicast
      g1[1] = (int)((unsigned)DIM << 16);                // tensor_dim0[15:0]
      g1[2] = (TBLK << 16);                              // tensor_dim1[15:0]
      g1[3] = (int)((unsigned)DIM << 16);                // tile_dim0
      g1[4] = TBLK;                                      // tile_dim1 (tile_dim2=0)
      g1[5] = DIM;                                       // tensor_dim0_stride lo
      g1[6] = 0;
      g1[7] = 0;
      v4i g2 = {0, 0, 0, 0};
      v4i g3 = {0, 0, 0, 0};
      v8i g4 = {0, 0, 0, 0, 0, 0, 0, 0};
      __builtin_amdgcn_tensor_load_to_lds(g0, g1, g2, g3, g4, 0);
      __builtin_amdgcn_s_wait_tensorcnt(0);
    }
    // warm GL2 for next block's keys/values (512 threads cover 8 rows x 4KB)
    if (blk + 1 < NBLK) {
      const size_t nb = (size_t)(t0 + TBLK) * DIM + (size_t)(tid >> 6) * DIM + (size_t)(tid & 63) * 16;
      __builtin_prefetch(kb + nb, 0, 1);
      __builtin_prefetch(vb + nb, 0, 1);
    }
    __syncthreads();

    // ---- GEMM phase (f32 WMMA) ----
    // dots0 = Kblk (16x1024, rows>=8 are harmless duplicates) * C^T  (waves 0..7)
    // G     = Kblk * Kblk^T                                          (waves 8..9)
    // Garbage confinement: dup A rows pollute only unused D rows 8..15; dup B cols
    // (Gram) pollute only unused G cols 8..15 -> branch-free inner loops.
    if (wv < 10) {
      const int m16 = lane & 15;
      const int hi  = lane >> 4;                         // K-pair select 0/1
      const int kp  = wv & 1;                            // 512-dim K partition
      const float* aRow = Kst + (lane & 7) * DIM + 2 * hi;
      v8f acc = {0.f, 0.f, 0.f, 0.f, 0.f, 0.f, 0.f, 0.f};
      const int c0 = kp * 128;
      if (wv < 8) {
        const int tile = wv >> 1;                        // centroid tile 0..3
        const float* bRow = C + (tile * 16 + m16) * DIM + hi;
#pragma unroll 4
        for (int c = c0; c < c0 + 128; ++c) {
          v2f a = *(const v2f*)(aRow + 4 * c);           // ds_load_b64
          v2f bf; bf[0] = bRow[4 * c]; bf[1] = bRow[4 * c + 2];
          acc = __builtin_amdgcn_wmma_f32_16x16x4_f32(false, a, false, bf,
                                                      (short)0, acc, false, false);
        }
      } else {
        const float* bRow = Kst + (lane & 7) * DIM + hi; // Gram: B from key tile
#pragma unroll 4
        for (int c = c0; c < c0 + 128; ++c) {
          v2f a = *(const v2f*)(aRow + 4 * c);
          v2f bf; bf[0] = bRow[4 * c]; bf[1] = bRow[4 * c + 2];
          acc = __builtin_amdgcn_wmma_f32_16x16x4_f32(false, a, false, bf,
                                                      (short)0, acc, false, false);
        }
      }
      float* pp = P + (size_t)(wv * 32 + lane) * 8;
#pragma unroll
      for (int i = 0; i < 8; ++i) pp[i] = acc[i];
    }
    __syncthreads();

    // ---- reduce the two K-partitions; scatter D tiles to dots / G ----
    if (wv < 5) {
      const int m16 = lane & 15;
      const int hi  = lane >> 4;
      const float* p0 = P + (size_t)((wv * 2 + 0) * 32 + lane) * 8;
      const float* p1 = P + (size_t)((wv * 2 + 1) * 32 + lane) * 8;
#pragma unroll
      for (int i = 0; i < 8; ++i) {
        const float s = p0[i] + p1[i];
        const int m = i + 8 * hi;                 // token row 0..15
        if (wv < 4) dots[m * MST + wv * 16 + m16] = s;
        else        G[m * 16 + m16] = s;
      }
    }
    __syncthreads();

    // ---- sequential per-token decisions (wave 0; 2 state slots per lane) ----
    if (wv == 0) {
      for (int t = 0; t < TBLK; ++t) {
        const float kk   = G[t * 16 + t];                 // ||k_t||^2
        const float invk = __frsqrt_rn(kk + EPS_F);
        const int r0 = lane, r1 = lane + 32;
        const float d0 = dots[t * MST + r0];
        const float d1 = dots[t * MST + r1];
        const float s0 = (cnt[r0] > 0.0f) ? d0 * ivc[r0] * invk : NEGINF;
        const float s1 = (cnt[r1] > 0.0f) ? d1 * ivc[r1] * invk : NEGINF;
        float bv; int bi;
        if (s1 > s0) { bv = s1; bi = r1; } else { bv = s0; bi = r0; }
#pragma unroll
        for (int off = 1; off < 32; off <<= 1) {          // first-index argmax
          const float ov = __shfl_xor(bv, off, 32);
          const int   oi = __shfl_xor(bi, off, 32);
          if (ov > bv || (ov == bv && oi < bi)) { bv = ov; bi = oi; }
        }
        const int   Kc    = *Kp;
        const int   docr  = (Kc == 0) || (bv < THRESH_F && Kc < MST);
        const int   slot  = docr ? Kc : bi;               // create => counts[slot]==0
        const float denom = cnt[slot] + 1.0f;
        const float rd    = 1.0f / denom;                 // create: rd==1 kills old term
        const float omrd  = 1.0f - rd;
        // rank-1 fixup of remaining tokens' dots against the updated centroid
        if (lane > t && lane < TBLK) {
          const float dv = dots[lane * MST + slot];
          dots[lane * MST + slot] = dv * omrd + G[t * 16 + lane] * rd;
        }
        if (lane == 0) {
          const float dct = dots[t * MST + slot];         // c.k_t (pre-update)
          const float ncn = ncs[slot] * omrd * omrd + 2.0f * omrd * rd * dct + kk * rd * rd;
          ncs[slot]  = ncn;
          ivc[slot]  = __frsqrt_rn(ncn + EPS_F);
          cnt[slot]  = denom;
          slotA[t]   = slot;
          denA[t]    = rd;
          *Kp        = Kc + docr;
        }
        __builtin_amdgcn_wave_barrier();
      }
    }
    __syncthreads();

    // ---- vector updates: thread owns 2 columns => per-address program order ----
    {
      const int j0 = tid * 2;
      for (int t = 0; t < TBLK; ++t) {
        const int   slot = slotA[t];
        const float rd   = denA[t];
        const float omrd = 1.0f - rd;
        // centroid row (LDS), keys from TDM-staged tile
        float* cp = C + slot * DIM + j0;
        const float* kp2 = Kst + t * DIM + j0;
        cp[0] = cp[0] * omrd + kp2[0] * rd;
        cp[1] = cp[1] * omrd + kp2[1] * rd;
        // state row (global, in-place in d_out)
        const float* vp = vb + (size_t)(t0 + t) * DIM + j0;
        float* sp = ob + slot * DIM + j0;
        sp[0] = sp[0] * omrd + vp[0] * rd;
        sp[1] = sp[1] * omrd + vp[1] * rd;
      }
    }
    __syncthreads();
  }
}

extern "C" void kernel_launch(void* const* d_in, const int* in_sizes, int n_in,
                              void* d_out, int out_size, void* d_ws, size_t ws_size,
                              hipStream_t stream) {
  const float* keys = (const float*)d_in[0];
  const float* vals = (const float*)d_in[1];
  float* out = (float*)d_out;
  const int B = in_sizes[0] / (SEQ * DIM);   // 8
  const size_t shmem = (size_t)LDS_FLOATS * sizeof(float);  // ~310 KB of the 320 KB WGP LDS
  hipLaunchKernelGGL(sparse_state_agg_kernel, dim3(B), dim3(NTHREADS), shmem, stream,
                     keys, vals, out);
}